// FraudGraphSAGE_90159953477680
// MI455X (gfx1250) — compile-verified
//
#include <hip/hip_runtime.h>

#define N_NODES 100000
#define N_EDGES 800000
#define IN_F    64
#define HID     128
#define OUT_F   64
#define BN_EPS  1e-5f

// workspace layout (in floats)
#define OFF_DEG   0
#define OFF_AGG1  (N_NODES)                          // deg/invdeg: N
#define OFF_H     (OFF_AGG1 + N_NODES * IN_F)        // agg1: N*64
#define OFF_AGG2  (OFF_H    + N_NODES * HID)         // h: N*128
#define OFF_STATS (OFF_AGG2 + N_NODES * HID)         // agg2: N*128
#define WS_FLOATS (OFF_STATS + 512)                  // sum[128],sq[128],scale[128],shift[128]

typedef float v2f __attribute__((ext_vector_type(2)));
typedef float v8f __attribute__((ext_vector_type(8)));

__device__ __forceinline__ void atomicAddF(float* p, float v) {
    // lowers to global_atomic_add_f32 (no CAS loop)
    unsafeAtomicAdd(p, v);
}

// ---------------------------------------------------------------- zero
__global__ __launch_bounds__(256) void zero_ws_kernel(float4* __restrict__ p, int n4) {
    int i = blockIdx.x * 256 + threadIdx.x;
    if (i < n4) p[i] = make_float4(0.f, 0.f, 0.f, 0.f);
}

// ---------------------------------------------------------------- degree
__global__ __launch_bounds__(256) void degree_kernel(const int* __restrict__ edst,
                                                     float* __restrict__ deg) {
    int e = blockIdx.x * 256 + threadIdx.x;
    if (e < N_EDGES) atomicAddF(&deg[edst[e]], 1.0f);
}

__global__ __launch_bounds__(256) void invdeg_kernel(float* __restrict__ deg) {
    int i = blockIdx.x * 256 + threadIdx.x;
    if (i < N_NODES) deg[i] = 1.0f / fmaxf(deg[i], 1.0f);
}

// ---------------------------------------------------------------- edge scatter-add
template <int F>
__global__ __launch_bounds__(256) void aggregate_kernel(const float* __restrict__ feat,
                                                        const int* __restrict__ esrc,
                                                        const int* __restrict__ edst,
                                                        float* __restrict__ agg) {
    const int CH = F / 4;                     // float4 chunks per row
    int tid = blockIdx.x * 256 + threadIdx.x; // < E * CH  (max 25.6M)
    int e = tid / CH;
    if (e >= N_EDGES) return;
    int c = (tid - e * CH) * 4;
    int s = esrc[e];
    int d = edst[e];
    const float4 v = *reinterpret_cast<const float4*>(feat + (size_t)s * F + c);
    float* p = agg + (size_t)d * F + c;
    atomicAddF(p + 0, v.x);
    atomicAddF(p + 1, v.y);
    atomicAddF(p + 2, v.z);
    atomicAddF(p + 3, v.w);
}

// ---------------------------------------------------------------- layer-1 GEMM + BN stats
// h_pre = x @ Wself(64x128) + (agg1*invdeg) @ Wneigh(64x128) + b1
// one block = 16 rows x 128 cols; 8 waves, one 16x16 WMMA tile each
__global__ __launch_bounds__(256) void gemm1_kernel(const float* __restrict__ x,
                                                    const float* __restrict__ agg,
                                                    const float* __restrict__ invdeg,
                                                    const float* __restrict__ Wself,
                                                    const float* __restrict__ Wneigh,
                                                    const float* __restrict__ bias,
                                                    float* __restrict__ h,
                                                    float* __restrict__ stat_sum,
                                                    float* __restrict__ stat_sq) {
    __shared__ float lds[128 * 16];           // A-tile, K-major: lds[k*16 + m], K = 64(x) + 64(agg)
    const int row0 = blockIdx.x * 16;
    const int t = threadIdx.x;

    // cooperative stage: 2048 elements, coalesced along k
    #pragma unroll
    for (int i = 0; i < 8; ++i) {
        int g = t + i * 256;
        int m = g >> 7;                        // 0..15
        int k = g & 127;                       // 0..127
        int row = row0 + m;
        float v = (k < IN_F) ? x[(size_t)row * IN_F + k]
                             : agg[(size_t)row * IN_F + (k - IN_F)] * invdeg[row];
        lds[k * 16 + m] = v;
    }
    __syncthreads();

    const int wave = t >> 5;                   // 0..7 -> col tile
    const int lane = t & 31;
    const int m    = lane & 15;
    const int hi   = lane >> 4;                // 0/1
    const int n    = wave * 16 + m;

    v8f c = {};
    #pragma unroll
    for (int k0 = 0; k0 < IN_F; k0 += 4) {     // self term
        int ka = k0 + 2 * hi;
        v2f a, b;
        a.x = lds[(ka + 0) * 16 + m];
        a.y = lds[(ka + 1) * 16 + m];
        b.x = Wself[(size_t)(ka + 0) * HID + n];
        b.y = Wself[(size_t)(ka + 1) * HID + n];
        c = __builtin_amdgcn_wmma_f32_16x16x4_f32(false, a, false, b, (short)0, c, false, false);
    }
    #pragma unroll
    for (int k0 = 0; k0 < IN_F; k0 += 4) {     // neighbor term
        int ka = k0 + 2 * hi;
        v2f a, b;
        a.x = lds[(IN_F + ka + 0) * 16 + m];
        a.y = lds[(IN_F + ka + 1) * 16 + m];
        b.x = Wneigh[(size_t)(ka + 0) * HID + n];
        b.y = Wneigh[(size_t)(ka + 1) * HID + n];
        c = __builtin_amdgcn_wmma_f32_16x16x4_f32(false, a, false, b, (short)0, c, false, false);
    }

    const float bn = bias[n];
    float s = 0.f, q = 0.f;
    #pragma unroll
    for (int r = 0; r < 8; ++r) {
        float v = c[r] + bn;
        h[(size_t)(row0 + r + 8 * hi) * HID + n] = v;
        s += v;
        q += v * v;
    }
    // column-wise partial sums for batchnorm: combine lane & lane+16 (same n)
    s += __shfl_down(s, 16, 32);
    q += __shfl_down(q, 16, 32);
    if (hi == 0) {
        atomicAddF(&stat_sum[n], s);
        atomicAddF(&stat_sq[n], q);
    }
}

// ---------------------------------------------------------------- BN finalize: scale/shift
__global__ __launch_bounds__(128) void bn_finalize_kernel(const float* __restrict__ stat_sum,
                                                          const float* __restrict__ stat_sq,
                                                          const float* __restrict__ gamma,
                                                          const float* __restrict__ beta,
                                                          float* __restrict__ scale,
                                                          float* __restrict__ shift) {
    int f = threadIdx.x;                       // 0..127
    const float invN = 1.0f / (float)N_NODES;
    float mu  = stat_sum[f] * invN;
    float var = stat_sq[f] * invN - mu * mu;
    float inv = rsqrtf(var + BN_EPS);
    float g   = gamma[f];
    scale[f] = g * inv;
    shift[f] = beta[f] - g * inv * mu;
}

// ---------------------------------------------------------------- BN apply + ReLU (in place)
__global__ __launch_bounds__(256) void bn_relu_kernel(float4* __restrict__ h4,
                                                      const float4* __restrict__ scale4,
                                                      const float4* __restrict__ shift4) {
    int i = blockIdx.x * 256 + threadIdx.x;    // < N*128/4
    if (i >= N_NODES * HID / 4) return;
    int cb = i & (HID / 4 - 1);                // float4 column chunk within row
    float4 v  = h4[i];
    float4 sc = scale4[cb];
    float4 sh = shift4[cb];
    v.x = fmaxf(fmaf(v.x, sc.x, sh.x), 0.f);
    v.y = fmaxf(fmaf(v.y, sc.y, sh.y), 0.f);
    v.z = fmaxf(fmaf(v.z, sc.z, sh.z), 0.f);
    v.w = fmaxf(fmaf(v.w, sc.w, sh.w), 0.f);
    h4[i] = v;
}

// ---------------------------------------------------------------- layer-2 GEMM
// out = h @ Wself2(128x64) + (agg2*invdeg) @ Wneigh2(128x64) + b2
// one block = 16 rows x 64 cols; 4 waves
__global__ __launch_bounds__(128) void gemm2_kernel(const float* __restrict__ h,
                                                    const float* __restrict__ agg,
                                                    const float* __restrict__ invdeg,
                                                    const float* __restrict__ Wself,
                                                    const float* __restrict__ Wneigh,
                                                    const float* __restrict__ bias,
                                                    float* __restrict__ out) {
    __shared__ float lds[256 * 16];            // 16 KB: K = 128(h) + 128(agg2)
    const int row0 = blockIdx.x * 16;
    const int t = threadIdx.x;                 // 0..127

    #pragma unroll
    for (int i = 0; i < 32; ++i) {
        int g = t + i * 128;
        int m = g >> 8;                        // 0..15
        int k = g & 255;                       // 0..255
        int row = row0 + m;
        float v = (k < HID) ? h[(size_t)row * HID + k]
                            : agg[(size_t)row * HID + (k - HID)] * invdeg[row];
        lds[k * 16 + m] = v;
    }
    __syncthreads();

    const int wave = t >> 5;                   // 0..3 -> col tile
    const int lane = t & 31;
    const int m    = lane & 15;
    const int hi   = lane >> 4;
    const int n    = wave * 16 + m;

    v8f c = {};
    #pragma unroll
    for (int k0 = 0; k0 < HID; k0 += 4) {      // self term
        int ka = k0 + 2 * hi;
        v2f a, b;
        a.x = lds[(ka + 0) * 16 + m];
        a.y = lds[(ka + 1) * 16 + m];
        b.x = Wself[(size_t)(ka + 0) * OUT_F + n];
        b.y = Wself[(size_t)(ka + 1) * OUT_F + n];
        c = __builtin_amdgcn_wmma_f32_16x16x4_f32(false, a, false, b, (short)0, c, false, false);
    }
    #pragma unroll
    for (int k0 = 0; k0 < HID; k0 += 4) {      // neighbor term
        int ka = k0 + 2 * hi;
        v2f a, b;
        a.x = lds[(HID + ka + 0) * 16 + m];
        a.y = lds[(HID + ka + 1) * 16 + m];
        b.x = Wneigh[(size_t)(ka + 0) * OUT_F + n];
        b.y = Wneigh[(size_t)(ka + 1) * OUT_F + n];
        c = __builtin_amdgcn_wmma_f32_16x16x4_f32(false, a, false, b, (short)0, c, false, false);
    }

    const float bn = bias[n];
    #pragma unroll
    for (int r = 0; r < 8; ++r)
        out[(size_t)(row0 + r + 8 * hi) * OUT_F + n] = c[r] + bn;
}

// ---------------------------------------------------------------- launch
extern "C" void kernel_launch(void* const* d_in, const int* in_sizes, int n_in,
                              void* d_out, int out_size, void* d_ws, size_t ws_size,
                              hipStream_t stream) {
    const float* x      = (const float*)d_in[0];
    const int*   esrc   = (const int*)d_in[1];
    const int*   edst   = (const int*)d_in[2];
    const float* Wself1 = (const float*)d_in[3];
    const float* Wneigh1= (const float*)d_in[4];
    const float* b1     = (const float*)d_in[5];
    const float* gamma  = (const float*)d_in[6];
    const float* beta   = (const float*)d_in[7];
    const float* Wself2 = (const float*)d_in[8];
    const float* Wneigh2= (const float*)d_in[9];
    const float* b2     = (const float*)d_in[10];
    float* out = (float*)d_out;
    float* ws  = (float*)d_ws;

    float* invdeg = ws + OFF_DEG;
    float* agg1   = ws + OFF_AGG1;
    float* h      = ws + OFF_H;
    float* agg2   = ws + OFF_AGG2;
    float* ssum   = ws + OFF_STATS;
    float* ssq    = ssum + 128;
    float* sscale = ssum + 256;
    float* sshift = ssum + 384;

    // 1. zero scratch (deg, agg1, agg2, stats; h is fully overwritten)
    {
        int n4 = WS_FLOATS / 4;
        zero_ws_kernel<<<(n4 + 255) / 256, 256, 0, stream>>>((float4*)ws, n4);
    }
    // 2. in-degree
    degree_kernel<<<(N_EDGES + 255) / 256, 256, 0, stream>>>(edst, invdeg);
    // 3. layer-1 neighbor sum (F=64)
    {
        int threads = N_EDGES * (IN_F / 4);
        aggregate_kernel<IN_F><<<(threads + 255) / 256, 256, 0, stream>>>(x, esrc, edst, agg1);
    }
    // 4. invdeg = 1/max(deg,1)
    invdeg_kernel<<<(N_NODES + 255) / 256, 256, 0, stream>>>(invdeg);
    // 5. layer-1 GEMM (WMMA f32) + BN partial stats
    gemm1_kernel<<<N_NODES / 16, 256, 0, stream>>>(x, agg1, invdeg, Wself1, Wneigh1, b1,
                                                   h, ssum, ssq);
    // 6. BN scale/shift
    bn_finalize_kernel<<<1, 128, 0, stream>>>(ssum, ssq, gamma, beta, sscale, sshift);
    // 7. BN apply + ReLU
    {
        int n4 = N_NODES * HID / 4;
        bn_relu_kernel<<<(n4 + 255) / 256, 256, 0, stream>>>((float4*)h, (const float4*)sscale,
                                                             (const float4*)sshift);
    }
    // 8. layer-2 neighbor sum (F=128)
    {
        int threads = N_EDGES * (HID / 4);
        aggregate_kernel<HID><<<(threads + 255) / 256, 256, 0, stream>>>(h, esrc, edst, agg2);
    }
    // 9. layer-2 GEMM (WMMA f32) -> d_out
    gemm2_kernel<<<N_NODES / 16, 128, 0, stream>>>(h, agg2, invdeg, Wself2, Wneigh2, b2, out);
}